// MoEFNO_15032385536123
// MI455X (gfx1250) — compile-verified
//
#include <hip/hip_runtime.h>
#include <hip/hip_bf16.h>

// ---------------- problem constants ----------------
constexpr int E_ = 8, NL_ = 4, CH_ = 32, M1_ = 16, M2_ = 16;
constexpr int HD_ = 64, NH_ = 4, NENC_ = 2;
constexpr int B_ = 32, H_ = 128, W_ = 128, COUT_ = 1, PCH_ = 128;
constexpr int HW_ = H_ * W_;     // 16384
constexpr int KY_ = 16;          // kept ky modes
constexpr int KX32_ = 32;        // kept kx modes (0..15, 112..127)

typedef float v2f __attribute__((ext_vector_type(2)));
typedef float v8f __attribute__((ext_vector_type(8)));

// ---------------- WMMA helpers (V_WMMA_F32_16X16X4_F32) ----------------
__device__ __forceinline__ v8f wmma4(v2f a, v2f b, v8f c) {
  // 8 args: (neg_a, A, neg_b, B, c_mod, C, reuse_a, reuse_b)
  return __builtin_amdgcn_wmma_f32_16x16x4_f32(false, a, false, b, (short)0, c,
                                               false, false);
}
__device__ __forceinline__ v8f v8zero() {
  v8f z = {0.f, 0.f, 0.f, 0.f, 0.f, 0.f, 0.f, 0.f};
  return z;
}
// A tile (MxK, row-major, leading dim lda): lanes 0-15 -> K {k0,k0+1}, 16-31 -> {k0+2,k0+3}
__device__ __forceinline__ v2f ldA(const float* A, int lda, int row0, int k0,
                                   int mr, int kh) {
  const float* p = A + (size_t)(row0 + mr) * lda + k0 + 2 * kh;
  v2f a; a.x = p[0]; a.y = p[1];
  return a;
}
// B tile (KxN, row-major, leading dim ldb)
__device__ __forceinline__ v2f ldB(const float* Bm, int ldb, int k0, int col0,
                                   int mr, int kh) {
  const float* p = Bm + (size_t)(k0 + 2 * kh) * ldb + col0 + mr;
  v2f b; b.x = p[0]; b.y = p[ldb];
  return b;
}
// C/D tile store: VGPR r -> rows r (lanes 0-15) and r+8 (lanes 16-31)
__device__ __forceinline__ void stC(float* C, int ldc, int row0, int col0,
                                    int mr, int kh, v8f c) {
#pragma unroll
  for (int r = 0; r < 8; ++r)
    C[(size_t)(row0 + r + 8 * kh) * ldc + col0 + mr] = c[r];
}

__device__ __forceinline__ float gelu_tanh(float x) {
  const float c0 = 0.7978845608028654f;  // sqrt(2/pi)
  float t = tanhf(c0 * (x + 0.044715f * x * x * x));
  return 0.5f * x * (1.f + t);
}

// ---------------- twiddle-table init ----------------
__global__ void k_init_tables(float* Fwc, float* Fws, float* Fhr, float* Fhi,
                              float* Ghr, float* Ghi, float* IA, float* IB) {
  const float TWO_PI = 6.283185307179586f;
  int stride = gridDim.x * blockDim.x;
  int t0 = blockIdx.x * blockDim.x + threadIdx.x;
  // W-forward: [w][ky]  X = sum_w v * e^{-i 2pi w ky /128}
  for (int i = t0; i < 128 * KY_; i += stride) {
    int w = i >> 4, ky = i & 15;
    float ang = TWO_PI * (float)((w * ky) & 127) / 128.f;
    Fwc[i] = cosf(ang);
    Fws[i] = -sinf(ang);
  }
  // H-forward: [kx32][h], kx = kx32<16 ? kx32 : kx32+96
  for (int i = t0; i < KX32_ * 128; i += stride) {
    int kx = i >> 7, h = i & 127;
    int KX = (kx < 16) ? kx : kx + 96;
    float ang = TWO_PI * (float)((h * KX) & 127) / 128.f;
    Fhr[i] = cosf(ang);
    Fhi[i] = -sinf(ang);
  }
  // H-inverse: [h][kx32], e^{+i...}
  for (int i = t0; i < 128 * KX32_; i += stride) {
    int h = i >> 5, kx = i & 31;
    int KX = (kx < 16) ? kx : kx + 96;
    float ang = TWO_PI * (float)((h * KX) & 127) / 128.f;
    Ghr[i] = cosf(ang);
    Ghi[i] = sinf(ang);
  }
  // W-inverse: [ky][w], Hermitian weights c_ky and 1/(H*W) folded in
  for (int i = t0; i < KY_ * 128; i += stride) {
    int ky = i >> 7, w = i & 127;
    float c = (ky == 0) ? 1.f : 2.f;
    float ang = TWO_PI * (float)((ky * w) & 127) / 128.f;
    IA[i] = c * cosf(ang) * (1.f / 16384.f);
    IB[i] = c * sinf(ang) * (1.f / 16384.f);
  }
}

// ---------------- router (one block per batch, thread = seq position) ----------------
__device__ __forceinline__ void ln64(const float* t, float* o, const float* s,
                                     const float* b) {
  float mu = 0.f;
#pragma unroll
  for (int d = 0; d < HD_; ++d) mu += t[d];
  mu *= (1.f / HD_);
  float var = 0.f;
#pragma unroll
  for (int d = 0; d < HD_; ++d) { float z = t[d] - mu; var += z * z; }
  var *= (1.f / HD_);
  float inv = rsqrtf(var + 1e-5f);
#pragma unroll
  for (int d = 0; d < HD_; ++d) o[d] = (t[d] - mu) * inv * s[d] + b[d];
}

__global__ __launch_bounds__(128) void k_router(
    const float* x, const float* enc_in_w, const float* enc_in_b,
    const float* qkv_w, const float* qkv_b, const float* ao_w,
    const float* ao_b, const float* ln1_s, const float* ln1_b,
    const float* ff1_w, const float* ff1_b, const float* ff2_w,
    const float* ff2_b, const float* ln2_s, const float* ln2_b,
    const float* fc_w, const float* fc_b, int* idx_out) {
  __shared__ float kbuf[128 * HD_];  // 32 KB
  __shared__ float vbuf[128 * HD_];  // 32 KB
  int b = blockIdx.x;
  int s = threadIdx.x;  // 0..127 (S = W = 128)

  float h[HD_];
  float ic = x[(size_t)b * HW_ + s];  // x[b,0,0,s]
#pragma unroll
  for (int d = 0; d < HD_; ++d) h[d] = ic * enc_in_w[d] + enc_in_b[d];

  for (int l = 0; l < NENC_; ++l) {
    const float* Wqkv = qkv_w + (size_t)l * HD_ * 3 * HD_;
    const float* bqkv = qkv_b + (size_t)l * 3 * HD_;
    float q[HD_];
#pragma unroll 4
    for (int j = 0; j < HD_; ++j) {
      float aq = bqkv[j], ak = bqkv[HD_ + j], av = bqkv[2 * HD_ + j];
#pragma unroll
      for (int d = 0; d < HD_; ++d) {
        float hv = h[d];
        const float* wr = Wqkv + (size_t)d * 3 * HD_;
        aq += hv * wr[j];
        ak += hv * wr[HD_ + j];
        av += hv * wr[2 * HD_ + j];
      }
      q[j] = aq;
      kbuf[s * HD_ + j] = ak;
      vbuf[s * HD_ + j] = av;
    }
    __syncthreads();

    float o[HD_];
    const float scale = 0.25f;  // 1/sqrt(dh), dh = 16
#pragma unroll
    for (int hh = 0; hh < NH_; ++hh) {
      int d0 = hh * 16;
      float m = -1e30f;
      for (int kk = 0; kk < 128; ++kk) {
        float dsc = 0.f;
#pragma unroll
        for (int d = 0; d < 16; ++d) dsc += q[d0 + d] * kbuf[kk * HD_ + d0 + d];
        m = fmaxf(m, dsc * scale);
      }
      float denom = 0.f;
      float acc[16];
#pragma unroll
      for (int d = 0; d < 16; ++d) acc[d] = 0.f;
      for (int kk = 0; kk < 128; ++kk) {
        float dsc = 0.f;
#pragma unroll
        for (int d = 0; d < 16; ++d) dsc += q[d0 + d] * kbuf[kk * HD_ + d0 + d];
        float e = __expf(dsc * scale - m);
        denom += e;
#pragma unroll
        for (int d = 0; d < 16; ++d) acc[d] += e * vbuf[kk * HD_ + d0 + d];
      }
      float inv = 1.f / denom;
#pragma unroll
      for (int d = 0; d < 16; ++d) o[d0 + d] = acc[d] * inv;
    }
    __syncthreads();  // done reading kbuf/vbuf for this layer

    // h = ln1(h + o @ ao_w + ao_b)
    const float* Wa = ao_w + (size_t)l * HD_ * HD_;
    float t64[HD_];
#pragma unroll 4
    for (int d = 0; d < HD_; ++d) {
      float a = ao_b[l * HD_ + d];
#pragma unroll
      for (int j = 0; j < HD_; ++j) a += o[j] * Wa[j * HD_ + d];
      t64[d] = h[d] + a;
    }
    ln64(t64, h, ln1_s + l * HD_, ln1_b + l * HD_);

    // ff + ln2
    const float* W1 = ff1_w + (size_t)l * HD_ * 4 * HD_;
    const float* b1 = ff1_b + (size_t)l * 4 * HD_;
    const float* W2 = ff2_w + (size_t)l * 4 * HD_ * HD_;
    const float* b2 = ff2_b + (size_t)l * HD_;
    float f64[HD_];
#pragma unroll
    for (int d = 0; d < HD_; ++d) f64[d] = b2[d];
    for (int j = 0; j < 4 * HD_; ++j) {
      float f = b1[j];
#pragma unroll
      for (int d = 0; d < HD_; ++d) f += h[d] * W1[d * 4 * HD_ + j];
      f = fmaxf(f, 0.f);
      const float* w2r = W2 + (size_t)j * HD_;
#pragma unroll
      for (int d = 0; d < HD_; ++d) f64[d] += f * w2r[d];
    }
#pragma unroll
    for (int d = 0; d < HD_; ++d) t64[d] = h[d] + f64[d];
    ln64(t64, h, ln2_s + l * HD_, ln2_b + l * HD_);
  }

  // feats = mean over s; logits; argmax
  __syncthreads();
#pragma unroll
  for (int d = 0; d < HD_; ++d) kbuf[s * HD_ + d] = h[d];
  __syncthreads();
  if (s < HD_) {
    float a = 0.f;
    for (int ss = 0; ss < 128; ++ss) a += kbuf[ss * HD_ + s];
    vbuf[s] = a * (1.f / 128.f);
  }
  __syncthreads();
  if (s == 0) {
    float best = -1e30f;
    int bi = 0;
    for (int e = 0; e < E_; ++e) {
      float lg = fc_b[e];
      for (int d = 0; d < HD_; ++d) lg += vbuf[d] * fc_w[d * E_ + e];
      if (lg > best) { best = lg; bi = e; }
    }
    idx_out[b] = bi;
  }
}

// ---------------- lift: v[b,d,hw] = x[b,0,hw]*lw[e,d] + lb[e,d] ----------------
__global__ __launch_bounds__(256) void k_lift(const float* x,
                                              const float* lift_w,
                                              const float* lift_b,
                                              const int* idx, float* v) {
  int b = blockIdx.y;
  int p = blockIdx.x * blockDim.x + threadIdx.x;
  int e = idx[b];
  float xv = x[(size_t)b * HW_ + p];
  const float* lw = lift_w + (size_t)e * CH_;
  const float* lb = lift_b + (size_t)e * CH_;
#pragma unroll 8
  for (int d = 0; d < CH_; ++d)
    v[((size_t)b * CH_ + d) * HW_ + p] = xv * lw[d] + lb[d];
}

// ---------------- W-forward partial DFT: X1 = V(.,128) x F(128,16) ----------------
__global__ __launch_bounds__(128) void k_wfft(const float* v, const float* Fwc,
                                              const float* Fws, float* X1r,
                                              float* X1i) {
  __shared__ float sC[128 * KY_], sS[128 * KY_];
  for (int i = threadIdx.x; i < 128 * KY_; i += 128) { sC[i] = Fwc[i]; sS[i] = Fws[i]; }
  __syncthreads();
  int wave = threadIdx.x >> 5, lane = threadIdx.x & 31;
  int mr = lane & 15, kh = lane >> 4;
  size_t row0 = ((size_t)blockIdx.x * 4 + wave) * 16;  // rows over B*CH*H
  const float* A = v + row0 * 128;
  v8f ar = v8zero(), ai = v8zero();
#pragma unroll
  for (int k = 0; k < 128; k += 4) {
    v2f a = ldA(A, 128, 0, k, mr, kh);
    ar = wmma4(a, ldB(sC, KY_, k, 0, mr, kh), ar);
    ai = wmma4(a, ldB(sS, KY_, k, 0, mr, kh), ai);
  }
  stC(X1r + row0 * KY_, KY_, 0, 0, mr, kh, ar);
  stC(X1i + row0 * KY_, KY_, 0, 0, mr, kh, ai);
}

// ---------------- H-forward DFT (complex): X2(32x16) = F(32x128) x X1(128x16) ----
__global__ __launch_bounds__(64) void k_hfft(const float* X1r, const float* X1i,
                                             const float* Fhr, const float* Fhi,
                                             float* X2r, float* X2i) {
  __shared__ float sXr[128 * KY_], sXi[128 * KY_];   // 8+8 KB
  __shared__ float sFr[KX32_ * 128], sFi[KX32_ * 128];  // 16+16 KB
  size_t img = blockIdx.x;  // b*CH + c
  const float* xr = X1r + img * 128 * KY_;
  const float* xi = X1i + img * 128 * KY_;
  for (int i = threadIdx.x; i < 128 * KY_; i += 64) { sXr[i] = xr[i]; sXi[i] = xi[i]; }
  for (int i = threadIdx.x; i < KX32_ * 128; i += 64) { sFr[i] = Fhr[i]; sFi[i] = Fhi[i]; }
  __syncthreads();
  int wave = threadIdx.x >> 5, lane = threadIdx.x & 31;
  int mr = lane & 15, kh = lane >> 4;
  int kx0 = wave * 16;
  v8f aP = v8zero(), aQ = v8zero(), aI = v8zero();
#pragma unroll
  for (int k = 0; k < 128; k += 4) {
    v2f fr = ldA(sFr, 128, kx0, k, mr, kh);
    v2f fi = ldA(sFi, 128, kx0, k, mr, kh);
    v2f br = ldB(sXr, KY_, k, 0, mr, kh);
    v2f bi = ldB(sXi, KY_, k, 0, mr, kh);
    aP = wmma4(fr, br, aP);
    aQ = wmma4(fi, bi, aQ);
    aI = wmma4(fr, bi, aI);
    aI = wmma4(fi, br, aI);
  }
  v8f re = aP - aQ;
  stC(X2r + img * KX32_ * KY_, KY_, kx0, 0, mr, kh, re);
  stC(X2i + img * KX32_ * KY_, KY_, kx0, 0, mr, kh, aI);
}

// ---------------- per-mode 32x32 complex channel mix ----------------
__global__ __launch_bounds__(256) void k_specmul(const float* X2r,
                                                 const float* X2i,
                                                 const float* spec_wr,
                                                 const float* spec_wi,
                                                 const int* idx, int l,
                                                 float* Yr, float* Yi) {
  int b = blockIdx.y;
  int kx = blockIdx.x;  // 0..31
  int e = idx[b];
  int corner = kx >> 4;
  int kxl = kx & 15;
  int ky = threadIdx.x & 15;
  int o0 = threadIdx.x >> 4;  // 0..15 (also handles o0+16)
  size_t wbase = ((((size_t)e * NL_ + l) * 2 + corner) * CH_) * CH_ * (M1_ * M2_) +
                 (size_t)kxl * M2_ + ky;
  const float* xrp = X2r + ((size_t)b * CH_) * KX32_ * KY_ + kx * KY_ + ky;
  const float* xip = X2i + ((size_t)b * CH_) * KX32_ * KY_ + kx * KY_ + ky;
  float or0 = 0.f, oi0 = 0.f, or1 = 0.f, oi1 = 0.f;
#pragma unroll 4
  for (int i = 0; i < CH_; ++i) {
    float xr = xrp[(size_t)i * KX32_ * KY_];
    float xi = xip[(size_t)i * KX32_ * KY_];
    size_t wi0 = wbase + ((size_t)i * CH_ + o0) * (M1_ * M2_);
    size_t wi1 = wbase + ((size_t)i * CH_ + o0 + 16) * (M1_ * M2_);
    float wr0 = spec_wr[wi0], wj0 = spec_wi[wi0];
    float wr1 = spec_wr[wi1], wj1 = spec_wi[wi1];
    or0 += xr * wr0 - xi * wj0;  oi0 += xr * wj0 + xi * wr0;
    or1 += xr * wr1 - xi * wj1;  oi1 += xr * wj1 + xi * wr1;
  }
  size_t ob0 = ((size_t)b * CH_ + o0) * KX32_ * KY_ + kx * KY_ + ky;
  size_t ob1 = ((size_t)b * CH_ + o0 + 16) * KX32_ * KY_ + kx * KY_ + ky;
  Yr[ob0] = or0; Yi[ob0] = oi0;
  Yr[ob1] = or1; Yi[ob1] = oi1;
}

// ---------------- H-inverse: Z(128x16) = G(128x32) x Y(32x16) (complex) --------
__global__ __launch_bounds__(256) void k_hifft(const float* Yr, const float* Yi,
                                               const float* Ghr,
                                               const float* Ghi, float* Zr,
                                               float* Zi) {
  __shared__ float sYr[KX32_ * KY_], sYi[KX32_ * KY_];   // 2+2 KB
  __shared__ float sGr[128 * KX32_], sGi[128 * KX32_];   // 16+16 KB
  size_t img = blockIdx.x;  // b*CH + o
  const float* yr = Yr + img * KX32_ * KY_;
  const float* yi = Yi + img * KX32_ * KY_;
  for (int i = threadIdx.x; i < KX32_ * KY_; i += 256) { sYr[i] = yr[i]; sYi[i] = yi[i]; }
  for (int i = threadIdx.x; i < 128 * KX32_; i += 256) { sGr[i] = Ghr[i]; sGi[i] = Ghi[i]; }
  __syncthreads();
  int wave = threadIdx.x >> 5, lane = threadIdx.x & 31;
  int mr = lane & 15, kh = lane >> 4;
  int h0 = wave * 16;
  v8f aP = v8zero(), aQ = v8zero(), aI = v8zero();
#pragma unroll
  for (int k = 0; k < KX32_; k += 4) {
    v2f gr = ldA(sGr, KX32_, h0, k, mr, kh);
    v2f gi = ldA(sGi, KX32_, h0, k, mr, kh);
    v2f br = ldB(sYr, KY_, k, 0, mr, kh);
    v2f bi = ldB(sYi, KY_, k, 0, mr, kh);
    aP = wmma4(gr, br, aP);
    aQ = wmma4(gi, bi, aQ);
    aI = wmma4(gr, bi, aI);
    aI = wmma4(gi, br, aI);
  }
  v8f re = aP - aQ;
  stC(Zr + img * 128 * KY_, KY_, h0, 0, mr, kh, re);
  stC(Zi + img * 128 * KY_, KY_, h0, 0, mr, kh, aI);
}

// ---------------- W-inverse: spec(128x128) = Zr x IA - Zi x IB ----------------
__global__ __launch_bounds__(256) void k_wifft(const float* Zr, const float* Zi,
                                               const float* IA, const float* IB,
                                               float* spec) {
  __shared__ float sZr[128 * KY_], sZi[128 * KY_];  // 8+8 KB
  __shared__ float sA[KY_ * 128], sB[KY_ * 128];    // 8+8 KB
  size_t img = blockIdx.x;  // b*CH + o
  const float* zr = Zr + img * 128 * KY_;
  const float* zi = Zi + img * 128 * KY_;
  for (int i = threadIdx.x; i < 128 * KY_; i += 256) { sZr[i] = zr[i]; sZi[i] = zi[i]; }
  for (int i = threadIdx.x; i < KY_ * 128; i += 256) { sA[i] = IA[i]; sB[i] = IB[i]; }
  __syncthreads();
  int wave = threadIdx.x >> 5, lane = threadIdx.x & 31;
  int mr = lane & 15, kh = lane >> 4;
  int h0 = wave * 16;
  float* out = spec + img * HW_;
  for (int w0 = 0; w0 < 128; w0 += 16) {
    v8f acc = v8zero(), acc2 = v8zero();
#pragma unroll
    for (int k = 0; k < KY_; k += 4) {
      v2f ar = ldA(sZr, KY_, h0, k, mr, kh);
      v2f ai = ldA(sZi, KY_, h0, k, mr, kh);
      acc = wmma4(ar, ldB(sA, 128, k, w0, mr, kh), acc);
      acc2 = wmma4(ai, ldB(sB, 128, k, w0, mr, kh), acc2);
    }
    v8f res = acc - acc2;
    stC(out, 128, h0, w0, mr, kh, res);
  }
}

// ---------------- skip 1x1 conv + spec add + bias (+gelu) ----------------
__global__ __launch_bounds__(256) void k_skip(const float* vin,
                                              const float* spec,
                                              const float* skip_w,
                                              const float* skip_b,
                                              const int* idx, int l,
                                              int apply_gelu, float* vout) {
  __shared__ float sW[CH_ * CH_];  // 4 KB
  int b = blockIdx.y;
  int e = idx[b];
  const float* wsrc = skip_w + (((size_t)e * NL_ + l) * CH_) * CH_;
  for (int i = threadIdx.x; i < CH_ * CH_; i += 256) sW[i] = wsrc[i];
  __syncthreads();
  int wave = threadIdx.x >> 5, lane = threadIdx.x & 31;
  int mr = lane & 15, kh = lane >> 4;
  size_t hw0 = ((size_t)blockIdx.x * 8 + wave) * 16;
  const float* vb = vin + (size_t)b * CH_ * HW_;
  v8f a0 = v8zero(), a1 = v8zero();
#pragma unroll
  for (int k = 0; k < CH_; k += 4) {
    v2f a;  // A[m][k] = vin[b][k][hw0+m] (coalesced across mr)
    a.x = vb[(size_t)(k + 2 * kh) * HW_ + hw0 + mr];
    a.y = vb[(size_t)(k + 2 * kh + 1) * HW_ + hw0 + mr];
    a0 = wmma4(a, ldB(sW, CH_, k, 0, mr, kh), a0);
    a1 = wmma4(a, ldB(sW, CH_, k, 16, mr, kh), a1);
  }
  const float* sb = skip_b + ((size_t)e * NL_ + l) * CH_;
  const float* sp = spec + (size_t)b * CH_ * HW_;
  float* vo = vout + (size_t)b * CH_ * HW_;
#pragma unroll
  for (int r = 0; r < 8; ++r) {
    size_t m = hw0 + r + 8 * kh;
    float v0 = a0[r] + sb[mr] + sp[(size_t)mr * HW_ + m];
    float v1 = a1[r] + sb[16 + mr] + sp[(size_t)(16 + mr) * HW_ + m];
    if (apply_gelu) { v0 = gelu_tanh(v0); v1 = gelu_tanh(v1); }
    vo[(size_t)mr * HW_ + m] = v0;
    vo[(size_t)(16 + mr) * HW_ + m] = v1;
  }
}

// ---------------- fused p1 -> gelu -> p2 projection ----------------
__global__ __launch_bounds__(256) void k_p1p2(const float* vin,
                                              const float* p1_w,
                                              const float* p1_b,
                                              const float* p2_w,
                                              const float* p2_b,
                                              const int* idx, float* out) {
  __shared__ float sW1[CH_ * PCH_];  // 16 KB
  __shared__ float sB1[PCH_];
  __shared__ float sW2[PCH_];
  int b = blockIdx.y;
  int e = idx[b];
  const float* w1 = p1_w + (size_t)e * CH_ * PCH_;
  for (int i = threadIdx.x; i < CH_ * PCH_; i += 256) sW1[i] = w1[i];
  for (int i = threadIdx.x; i < PCH_; i += 256) {
    sB1[i] = p1_b[(size_t)e * PCH_ + i];
    sW2[i] = p2_w[(size_t)e * PCH_ + i];
  }
  __syncthreads();
  int wave = threadIdx.x >> 5, lane = threadIdx.x & 31;
  int mr = lane & 15, kh = lane >> 4;
  size_t hw0 = ((size_t)blockIdx.x * 8 + wave) * 16;
  const float* vb = vin + (size_t)b * CH_ * HW_;
  float partial[8];
#pragma unroll
  for (int r = 0; r < 8; ++r) partial[r] = 0.f;
  for (int p0 = 0; p0 < PCH_; p0 += 16) {
    v8f acc = v8zero();
#pragma unroll
    for (int k = 0; k < CH_; k += 4) {
      v2f a;
      a.x = vb[(size_t)(k + 2 * kh) * HW_ + hw0 + mr];
      a.y = vb[(size_t)(k + 2 * kh + 1) * HW_ + hw0 + mr];
      acc = wmma4(a, ldB(sW1, PCH_, k, p0, mr, kh), acc);
    }
    int p = p0 + mr;
    float w2 = sW2[p], b1v = sB1[p];
#pragma unroll
    for (int r = 0; r < 8; ++r)
      partial[r] += gelu_tanh(acc[r] + b1v) * w2;
  }
  // reduce the 16 N-lanes within each half-wave
#pragma unroll
  for (int off = 1; off < 16; off <<= 1)
#pragma unroll
    for (int r = 0; r < 8; ++r)
      partial[r] += __shfl_xor(partial[r], off, 32);
  float p2b = p2_b[(size_t)e * COUT_];
  if (mr == 0) {
#pragma unroll
    for (int r = 0; r < 8; ++r)
      out[(size_t)b * HW_ + hw0 + r + 8 * kh] = partial[r] + p2b;
  }
}

// ---------------- host orchestration ----------------
extern "C" void kernel_launch(void* const* d_in, const int* in_sizes, int n_in,
                              void* d_out, int out_size, void* d_ws,
                              size_t ws_size, hipStream_t stream) {
  const float* x = (const float*)d_in[0];
  const float* enc_in_w = (const float*)d_in[1];
  const float* enc_in_b = (const float*)d_in[2];
  const float* qkv_w = (const float*)d_in[3];
  const float* qkv_b = (const float*)d_in[4];
  const float* ao_w = (const float*)d_in[5];
  const float* ao_b = (const float*)d_in[6];
  const float* ln1_s = (const float*)d_in[7];
  const float* ln1_b = (const float*)d_in[8];
  const float* ff1_w = (const float*)d_in[9];
  const float* ff1_b = (const float*)d_in[10];
  const float* ff2_w = (const float*)d_in[11];
  const float* ff2_b = (const float*)d_in[12];
  const float* ln2_s = (const float*)d_in[13];
  const float* ln2_b = (const float*)d_in[14];
  const float* fc_w = (const float*)d_in[15];
  const float* fc_b = (const float*)d_in[16];
  const float* lift_w = (const float*)d_in[17];
  const float* lift_b = (const float*)d_in[18];
  const float* spec_wr = (const float*)d_in[19];
  const float* spec_wi = (const float*)d_in[20];
  const float* skip_w = (const float*)d_in[21];
  const float* skip_b = (const float*)d_in[22];
  const float* p1_w = (const float*)d_in[23];
  const float* p1_b = (const float*)d_in[24];
  const float* p2_w = (const float*)d_in[25];
  const float* p2_b = (const float*)d_in[26];

  char* ws = (char*)d_ws;
  size_t off = 0;
  auto alloc = [&](size_t bytes) -> void* {
    void* p = ws + off;
    off = (off + bytes + 255) & ~(size_t)255;
    return p;
  };
  int* idx = (int*)alloc(B_ * sizeof(int));
  float* Fwc = (float*)alloc(128 * KY_ * 4);
  float* Fws = (float*)alloc(128 * KY_ * 4);
  float* Fhr = (float*)alloc(KX32_ * 128 * 4);
  float* Fhi = (float*)alloc(KX32_ * 128 * 4);
  float* Ghr = (float*)alloc(128 * KX32_ * 4);
  float* Ghi = (float*)alloc(128 * KX32_ * 4);
  float* IA = (float*)alloc(KY_ * 128 * 4);
  float* IB = (float*)alloc(KY_ * 128 * 4);
  size_t vbytes = (size_t)B_ * CH_ * HW_ * 4;  // 64 MB
  float* vA = (float*)alloc(vbytes);
  float* vB = (float*)alloc(vbytes);
  float* specb = (float*)alloc(vbytes);
  size_t x1b = (size_t)B_ * CH_ * 128 * KY_ * 4;  // 8 MB
  float* X1r = (float*)alloc(x1b);
  float* X1i = (float*)alloc(x1b);
  size_t x2b = (size_t)B_ * CH_ * KX32_ * KY_ * 4;  // 2 MB
  float* X2r = (float*)alloc(x2b);
  float* X2i = (float*)alloc(x2b);
  float* Yr = (float*)alloc(x2b);
  float* Yi = (float*)alloc(x2b);
  float* Zr = (float*)alloc(x1b);
  float* Zi = (float*)alloc(x1b);

  k_init_tables<<<32, 256, 0, stream>>>(Fwc, Fws, Fhr, Fhi, Ghr, Ghi, IA, IB);
  k_router<<<B_, 128, 0, stream>>>(x, enc_in_w, enc_in_b, qkv_w, qkv_b, ao_w,
                                   ao_b, ln1_s, ln1_b, ff1_w, ff1_b, ff2_w,
                                   ff2_b, ln2_s, ln2_b, fc_w, fc_b, idx);
  k_lift<<<dim3(HW_ / 256, B_), 256, 0, stream>>>(x, lift_w, lift_b, idx, vA);

  float* vin = vA;
  float* vout = vB;
  for (int l = 0; l < NL_; ++l) {
    k_wfft<<<(B_ * CH_ * H_) / 64, 128, 0, stream>>>(vin, Fwc, Fws, X1r, X1i);
    k_hfft<<<B_ * CH_, 64, 0, stream>>>(X1r, X1i, Fhr, Fhi, X2r, X2i);
    k_specmul<<<dim3(KX32_, B_), 256, 0, stream>>>(X2r, X2i, spec_wr, spec_wi,
                                                   idx, l, Yr, Yi);
    k_hifft<<<B_ * CH_, 256, 0, stream>>>(Yr, Yi, Ghr, Ghi, Zr, Zi);
    k_wifft<<<B_ * CH_, 256, 0, stream>>>(Zr, Zi, IA, IB, specb);
    k_skip<<<dim3(HW_ / 128, B_), 256, 0, stream>>>(
        vin, specb, skip_w, skip_b, idx, l, (l < NL_ - 1) ? 1 : 0, vout);
    float* t = vin; vin = vout; vout = t;
  }
  k_p1p2<<<dim3(HW_ / 128, B_), 256, 0, stream>>>(vin, p1_w, p1_b, p2_w, p2_b,
                                                  idx, (float*)d_out);
}